// CRF_14379550507279
// MI455X (gfx1250) — compile-verified
//
#include <hip/hip_runtime.h>

typedef __attribute__((ext_vector_type(2))) float v2f;
typedef __attribute__((ext_vector_type(4))) float v4f;
typedef __attribute__((ext_vector_type(8))) float v8f;

#define CRF_B 512
#define CRF_S 2048
#define CRF_T 32

// ---------------------------------------------------------------------------
// Kernel A: masked emissions sum, accumulated through V_WMMA_F32_16X16X4_F32.
// B-matrix = all ones  ->  D[m,n] += sum_k A[m,k]; summing all D elements over
// all lanes gives 16 * (sum of all A inputs).  Exact f32 math; /16 is exact.
// grid 1024 x block 256 -> 262144 lanes; 8,388,608 float4s -> exactly 32
// iterations per lane, so EXEC is all-ones at every WMMA (required).
// ---------------------------------------------------------------------------
__global__ void __launch_bounds__(256) crf_emis_reduce(const float* __restrict__ emis,
                                                       const float* __restrict__ mask,
                                                       float* __restrict__ partials) {
    __shared__ float red[256];
    const int tid = threadIdx.x;
    const long long gid    = (long long)blockIdx.x * 256 + tid;
    const long long stride = (long long)gridDim.x * 256;                 // 262144
    const long long nvec   = (long long)CRF_B * CRF_S * CRF_T / 4;       // 8388608

    v8f c = {};                       // 16x16 f32 accumulator (8 VGPRs/lane)
    v2f ones; ones[0] = 1.0f; ones[1] = 1.0f;

    for (long long i = gid; i < nvec; i += stride) {
        // row r covers 8 float4s; s = r % 2048, b = r / 2048
        long long r = i >> 3;
        int       s = (int)(r & (CRF_S - 1));
        long long b = r >> 11;

        float m = mask[b * CRF_S + s];          // always a valid address
        float w = (s == 0) ? 1.0f : m;          // v_cndmask, no divergence

        v4f v = ((const v4f*)emis)[i];          // global_load_b128, streaming
        __builtin_prefetch(((const v4f*)emis) + i + stride, 0, 0); // speculative

        v2f a0; a0[0] = v[0] * w; a0[1] = v[1] * w;
        v2f a1; a1[0] = v[2] * w; a1[1] = v[3] * w;

        c = __builtin_amdgcn_wmma_f32_16x16x4_f32(false, a0, false, ones,
                                                  (short)0, c, false, false);
        c = __builtin_amdgcn_wmma_f32_16x16x4_f32(false, a1, false, ones,
                                                  (short)0, c, false, false);
    }

    float local = ((c[0] + c[1]) + (c[2] + c[3])) + ((c[4] + c[5]) + (c[6] + c[7]));
    red[tid] = local;
    __syncthreads();
    for (int off = 128; off > 0; off >>= 1) {
        if (tid < off) red[tid] += red[tid + off];
        __syncthreads();
    }
    if (tid == 0) partials[blockIdx.x] = red[0] * 0.0625f;   // undo the x16 (exact)
}

// ---------------------------------------------------------------------------
// Kernel B: transition terms, one block per batch b.
//   t1 = sum_t transitions[tags[b,0], t]                       (score0 part)
//   t2 = sum_{s>=1} transitions[tags[b,s-1], tags[b,s]] * mask[b,s]
//   partial = t1 + 32 * t2
// transitions (4KB) and the tags row (8KB) are staged in LDS.
// ---------------------------------------------------------------------------
__global__ void __launch_bounds__(256) crf_trans_reduce(const int* __restrict__ tags,
                                                        const float* __restrict__ mask,
                                                        const float* __restrict__ trans,
                                                        float* __restrict__ partials) {
    __shared__ float tl[CRF_T * CRF_T];
    __shared__ int   tg[CRF_S];
    __shared__ float red[256];
    const int tid = threadIdx.x;
    const int b   = blockIdx.x;

    for (int i = tid; i < CRF_T * CRF_T; i += 256) tl[i] = trans[i];
    for (int i = tid; i < CRF_S; i += 256)         tg[i] = tags[(long long)b * CRF_S + i];
    __syncthreads();

    float t1 = 0.0f;
    if (tid < CRF_T) t1 = tl[tg[0] * CRF_T + tid];

    float t2 = 0.0f;
    for (int s = tid + 1; s < CRF_S; s += 256) {   // coalesced mask reads
        t2 += tl[tg[s - 1] * CRF_T + tg[s]] * mask[(long long)b * CRF_S + s];
    }

    red[tid] = t1 + 32.0f * t2;                    // * num_tags
    __syncthreads();
    for (int off = 128; off > 0; off >>= 1) {
        if (tid < off) red[tid] += red[tid + off];
        __syncthreads();
    }
    if (tid == 0) partials[1024 + b] = red[0];
}

// ---------------------------------------------------------------------------
// Kernel C: deterministic fixed-order reduction of 1024 + 512 partials.
// ---------------------------------------------------------------------------
__global__ void __launch_bounds__(256) crf_finalize(const float* __restrict__ partials,
                                                    float* __restrict__ out) {
    __shared__ float red[256];
    const int tid = threadIdx.x;
    float local = 0.0f;
    for (int i = tid; i < 1536; i += 256) local += partials[i];
    red[tid] = local;
    __syncthreads();
    for (int off = 128; off > 0; off >>= 1) {
        if (tid < off) red[tid] += red[tid + off];
        __syncthreads();
    }
    if (tid == 0) out[0] = red[0];
}

extern "C" void kernel_launch(void* const* d_in, const int* in_sizes, int n_in,
                              void* d_out, int out_size, void* d_ws, size_t ws_size,
                              hipStream_t stream) {
    const float* emissions   = (const float*)d_in[0];   // (B,S,T) f32
    const int*   tags        = (const int*)  d_in[1];   // (B,S)   i32
    const float* mask        = (const float*)d_in[2];   // (B,S)   f32
    const float* transitions = (const float*)d_in[3];   // (T,T)   f32

    float* partials = (float*)d_ws;                     // 1536 floats used

    crf_emis_reduce <<<1024, 256, 0, stream>>>(emissions, mask, partials);
    crf_trans_reduce<<< 512, 256, 0, stream>>>(tags, mask, transitions, partials);
    crf_finalize    <<<   1, 256, 0, stream>>>(partials, (float*)d_out);
}